// JTNNVAE_81131932221573
// MI455X (gfx1250) — compile-verified
//
#include <hip/hip_runtime.h>
#include <hip/hip_bf16.h>
#include <math.h>

typedef float v2f __attribute__((ext_vector_type(2)));
typedef float v8f __attribute__((ext_vector_type(8)));

#define NB 1024          // molecules
#define LL 28            // latent
#define HH 448           // hidden
#define SEG_PER_MOL 20
#define NSEG (NB * SEG_PER_MOL)   // 20480 segments
#define NC 16            // candidates per segment

// ---------------------------------------------------------------------------
// Kernel 0: zero the two accumulators (loss-sum, correct-count) in workspace.
// ---------------------------------------------------------------------------
__global__ void init_accum(float* __restrict__ ws) {
    if (threadIdx.x == 0) { ws[0] = 0.0f; ws[1] = 0.0f; }
}

// ---------------------------------------------------------------------------
// Kernel 1: proj[1024x448] = X[1024x28] * W^T  (W is [448x28], Linear L->H)
// via V_WMMA_F32_16X16X4_F32, one wave per 16x16 output tile, 7 K-chunks.
//
// A (16x4 f32) lane layout: lanes 0-15 hold K={k0,k0+1}, lanes 16-31 K={k0+2,k0+3},
//   M = lane&15 (ISA 7.12.2 "32-bit A-Matrix 16x4").
// B (4x16) / C,D (16x16): rows striped across lanes within a VGPR.
// ---------------------------------------------------------------------------
__global__ __launch_bounds__(256) void proj_wmma(const float* __restrict__ X,
                                                 const float* __restrict__ W,
                                                 float* __restrict__ proj) {
    const int wave  = (blockIdx.x * blockDim.x + threadIdx.x) >> 5;
    const int lane  = threadIdx.x & 31;
    const int tileN = wave % (HH / 16);       // 28 N-tiles
    const int tileM = wave / (HH / 16);       // 64 M-tiles
    const int mrow  = tileM * 16 + (lane & 15);
    const int ncol  = tileN * 16 + (lane & 15);
    const int khi   = (lane >= 16) ? 2 : 0;

    v8f c = {};
#pragma unroll
    for (int k0 = 0; k0 < LL; k0 += 4) {      // 7 chunks of K=4
        const int kA = k0 + khi;
        v2f a, b;
        a.x = X[mrow * LL + kA];
        a.y = X[mrow * LL + kA + 1];
        // B[k][n] = W[n][k]  (proj = X * W^T)
        b.x = W[ncol * LL + kA];
        b.y = W[ncol * LL + kA + 1];
        c = __builtin_amdgcn_wmma_f32_16x16x4_f32(
                /*neg_a=*/false, a, /*neg_b=*/false, b,
                /*c_mod=*/(short)0, c, /*reuse_a=*/false, /*reuse_b=*/false);
    }

    // D 16x16 f32: VGPR r -> M = r (lanes 0-15) / r+8 (lanes 16-31), N = lane&15
    const int rbase = tileM * 16 + ((lane >= 16) ? 8 : 0);
    const int col   = tileN * 16 + (lane & 15);
#pragma unroll
    for (int r = 0; r < 8; ++r)
        proj[(size_t)(rbase + r) * HH + col] = c[r];
}

// ---------------------------------------------------------------------------
// Kernel 2: streaming scores + segmented log-softmax CE.
// One 512-thread block (16 wave32) per segment. Wave w computes the dot of
// candidate row (seg*16+w) with the segment's proj row (staged in LDS).
// HBM-bound: reads 587 MB of cand_vecs with coalesced b64 loads.
// ---------------------------------------------------------------------------
__global__ __launch_bounds__(512) void seg_ce(const float* __restrict__ proj,
                                              const float* __restrict__ cand,
                                              const int*   __restrict__ label_pos,
                                              float* __restrict__ accum) {
    __shared__ float pjs[HH];
    __shared__ float sc[NC];

    const int seg = blockIdx.x;
    const int mol = seg / SEG_PER_MOL;
    const int tid = threadIdx.x;

    // Stage the shared projection row (1.75 KB) into LDS.
    for (int i = tid; i < HH; i += 512) pjs[i] = proj[(size_t)mol * HH + i];
    __syncthreads();

    const int w = tid >> 5, lane = tid & 31;
    const float2* cp =
        (const float2*)(cand + (size_t)(seg * NC + w) * HH);   // 448 f32 = 224 f2

    float partial = 0.0f;
#pragma unroll
    for (int j = 0; j < 7; ++j) {             // 7 * 32 lanes * float2 = 448
        const float2 cv = cp[j * 32 + lane];  // coalesced 256B per wave per iter
        const int e = (j * 32 + lane) * 2;
        partial = fmaf(cv.x, pjs[e],     partial);
        partial = fmaf(cv.y, pjs[e + 1], partial);
    }
    // wave32 reduction
#pragma unroll
    for (int off = 16; off; off >>= 1)
        partial += __shfl_xor(partial, off, 32);
    if (lane == 0) sc[w] = partial;
    __syncthreads();

    // First wave: max / LSE / label over the 16 candidate scores.
    if (tid < 32) {
        const float s = (tid < NC) ? sc[tid] : -INFINITY;
        float m = s;
#pragma unroll
        for (int off = 16; off; off >>= 1)
            m = fmaxf(m, __shfl_xor(m, off, 32));
        float e = (tid < NC) ? __expf(s - m) : 0.0f;
#pragma unroll
        for (int off = 16; off; off >>= 1)
            e += __shfl_xor(e, off, 32);
        if (tid == 0) {
            const float lse  = __logf(e) + m;
            const int   loff = label_pos[seg] - seg * NC;
            const float lab  = sc[loff];
            atomicAdd(&accum[0], lse - lab);
            atomicAdd(&accum[1], (lab >= m) ? 1.0f : 0.0f);
        }
    }
}

// ---------------------------------------------------------------------------
// Kernel 3: finalize -> d_out = {loss_sum / n_trees, correct / n_seg}
// ---------------------------------------------------------------------------
__global__ void finalize(const float* __restrict__ accum, float* __restrict__ out) {
    if (threadIdx.x == 0) {
        out[0] = accum[0] / (float)NB;
        out[1] = accum[1] / (float)NSEG;
    }
}

extern "C" void kernel_launch(void* const* d_in, const int* in_sizes, int n_in,
                              void* d_out, int out_size, void* d_ws, size_t ws_size,
                              hipStream_t stream) {
    const float* X         = (const float*)d_in[0];   // [1024, 28]
    const float* W         = (const float*)d_in[1];   // [448, 28]
    const float* cand      = (const float*)d_in[2];   // [327680, 448]
    // d_in[3] batch_idx, d_in[4] segment_ids: derivable from indices, unused.
    const int*   label_pos = (const int*)d_in[5];     // [20480]

    float* ws   = (float*)d_ws;
    float* proj = ws + 16;                            // 64B-aligned, 1.79 MB used
    float* out  = (float*)d_out;

    init_accum<<<1, 32, 0, stream>>>(ws);

    const int nWaves  = (NB / 16) * (HH / 16);        // 64 * 28 = 1792 tiles
    const int nBlocks = nWaves / 8;                   // 8 waves per 256-thr block
    proj_wmma<<<nBlocks, 256, 0, stream>>>(X, W, proj);

    seg_ce<<<NSEG, 512, 0, stream>>>(proj, cand, label_pos, ws);

    finalize<<<1, 32, 0, stream>>>(ws, out);
}